// Net_62362925138472
// MI455X (gfx1250) — compile-verified
//
#include <hip/hip_runtime.h>
#include <math.h>

// ---------------------------------------------------------------------------
// GNN pipeline for MI455X (gfx1250, wave32).
// GEMMs: V_WMMA_F32_16X16X4_F32, K fixed at 64. Each wave preloads its full
// 64x16 B (weight) panel into 32 VGPRs (v2f bpan[16]) and sweeps 4 row tiles,
// so the unrolled inner loop is 16 x (global_load_b64 A + v_wmma) with zero
// redundant weight traffic. K=128 input linear = two accumulated K=64 passes.
// Edge propagation: float4 gathers + global_atomic_add_f32 scatters, all
// L2-resident (working set ~45MB << 192MB L2).
// ---------------------------------------------------------------------------

typedef __attribute__((ext_vector_type(2))) float v2f;
typedef __attribute__((ext_vector_type(8))) float v8f;

#define GNN_N 50000
#define GNN_E 800000
#define GNN_D 64
#define KSTEPS 16          // K = 64 = KSTEPS * 4
#define ROW_TILES 4        // row tiles (of 16) swept per wave

// ---------------------------------------------------------------------------
// C[M x Nc] (+)= A[M x 64] * B   with B[k][n] = Bsrc[k*bks + n*bns].
// initC=0: acc starts from bias (or 0);  initC=1: acc starts from C (for
// split-K accumulation).  relu applied at store.  M, Nc multiples of 16.
// One wave owns one 16-col panel and ROW_TILES row tiles.
// ---------------------------------------------------------------------------
__global__ __launch_bounds__(256) void gemm_wmma_k64_kernel(
    const float* __restrict__ A, int lda,
    const float* __restrict__ B, int bks, int bns,
    const float* __restrict__ bias, float* __restrict__ C,
    int M, int Nc, int initC, int relu)
{
    const int lane = threadIdx.x & 31;
    const int wave = threadIdx.x >> 5;
    const int tilesN = Nc >> 4;
    const int Mtiles = M >> 4;
    const int groups = (Mtiles + ROW_TILES - 1) / ROW_TILES;

    const int w = blockIdx.x * (blockDim.x >> 5) + wave;
    if (w >= groups * tilesN) return;          // wave-uniform exit
    const int tn = w % tilesN;
    const int g  = w / tilesN;

    const int r16 = lane & 15;                 // A row / B,C col within tile
    const int kb  = (lane >> 4) << 1;          // lanes 0-15 -> k+0, 16-31 -> k+2
    const int col0 = tn << 4;

    // ---- preload full 64x16 B panel into registers (32 VGPRs) ----
    const float* bcol = B + (size_t)(col0 + r16) * (size_t)bns;
    v2f bpan[KSTEPS];
#pragma unroll
    for (int ks = 0; ks < KSTEPS; ++ks) {
        const int k = ks * 4 + kb;
        bpan[ks].x = bcol[(size_t)k * (size_t)bks];
        bpan[ks].y = bcol[(size_t)(k + 1) * (size_t)bks];
    }

    const float bv = bias ? bias[col0 + r16] : 0.0f;
    const int mo = (lane >> 4) << 3;           // C row offset: 0 or 8

#pragma unroll
    for (int rt = 0; rt < ROW_TILES; ++rt) {
        const int tm = g * ROW_TILES + rt;     // wave-uniform
        if (tm >= Mtiles) break;
        const int row0 = tm << 4;

        float* crow = C + (size_t)(row0 + mo) * (size_t)Nc + col0 + r16;

        v8f acc;
        if (initC) {
#pragma unroll
            for (int r = 0; r < 8; ++r) acc[r] = crow[(size_t)r * (size_t)Nc];
        } else {
#pragma unroll
            for (int r = 0; r < 8; ++r) acc[r] = bv;
        }

        const float* arow = A + (size_t)(row0 + r16) * (size_t)lda + kb;
#pragma unroll
        for (int ks = 0; ks < KSTEPS; ++ks) {
            v2f a;
            a.x = arow[ks * 4];
            a.y = arow[ks * 4 + 1];
            acc = __builtin_amdgcn_wmma_f32_16x16x4_f32(
                false, a, false, bpan[ks], (short)0, acc, false, false);
        }

#pragma unroll
        for (int r = 0; r < 8; ++r) {
            float v = acc[r];
            if (relu) v = fmaxf(v, 0.0f);
            crow[(size_t)r * (size_t)Nc] = v;
        }
    }
}

// ---------------------------------------------------------------------------
// Degree / normalization helpers
// ---------------------------------------------------------------------------
__global__ void deg_init_kernel(float* __restrict__ deg, int n) {
    int i = blockIdx.x * blockDim.x + threadIdx.x;
    if (i < n) deg[i] = 1.0f;   // self-loop contributes 1
}

__global__ void deg_scatter_kernel(const int* __restrict__ dst,
                                   float* __restrict__ deg, int e) {
    int i = blockIdx.x * blockDim.x + threadIdx.x;
    if (i < e) atomicAdd(&deg[dst[i]], 1.0f);
}

__global__ void dinv_kernel(const float* __restrict__ deg,
                            float* __restrict__ dinv,
                            float* __restrict__ snorm, int n) {
    int i = blockIdx.x * blockDim.x + threadIdx.x;
    if (i < n) {
        float r = rsqrtf(deg[i]);
        dinv[i] = r;
        snorm[i] = r * r;       // self-loop norm = dinv^2
    }
}

__global__ void edge_norm_kernel(const int* __restrict__ src,
                                 const int* __restrict__ dst,
                                 const float* __restrict__ dinv,
                                 float* __restrict__ norm, int e) {
    int i = blockIdx.x * blockDim.x + threadIdx.x;
    if (i < e) norm[i] = dinv[src[i]] * dinv[dst[i]];
}

// out[n][d] = xw[n][d]*snorm[n] + bias[d]  (self-loop term + bias; full init
// of `out` before the atomic edge scatter)
__global__ void selfloop_bias_kernel(const float* __restrict__ xw,
                                     const float* __restrict__ snorm,
                                     const float* __restrict__ bias,
                                     float* __restrict__ out, int total) {
    int i = blockIdx.x * blockDim.x + threadIdx.x;
    if (i >= total) return;
    int node = i >> 6;
    int d = i & 63;
    out[i] = xw[i] * snorm[node] + bias[d];
}

__global__ void zero_kernel(float* __restrict__ p, int n) {
    int i = blockIdx.x * blockDim.x + threadIdx.x;
    if (i < n) p[i] = 0.0f;
}

// out[dst[e]][:] += xw[src[e]][:] * (norm ? norm[e] : 1)
// 16 threads per edge, float4 per thread -> 4 f32 atomics each; all
// L2-resident for this working set.
__global__ void scatter_edges_kernel(const float* __restrict__ xw,
                                     const int* __restrict__ src,
                                     const int* __restrict__ dst,
                                     const float* __restrict__ norm,
                                     float* __restrict__ out, int e) {
    int t = blockIdx.x * blockDim.x + threadIdx.x;
    if (t >= e * 16) return;
    int ed = t >> 4;
    int d = (t & 15) << 2;
    int s = src[ed];
    int q = dst[ed];
    float w = norm ? norm[ed] : 1.0f;
    const float4 v = *(const float4*)(xw + (size_t)s * GNN_D + d);
    float* o = out + (size_t)q * GNN_D + d;
    atomicAdd(o + 0, v.x * w);
    atomicAdd(o + 1, v.y * w);
    atomicAdd(o + 2, v.z * w);
    atomicAdd(o + 3, v.w * w);
}

// GRU gate math: gi/gh are [N x 192] (biases already folded in by the GEMM).
__global__ void gru_kernel(const float* __restrict__ gi,
                           const float* __restrict__ gh,
                           const float* __restrict__ h,
                           float* __restrict__ hn, int total) {
    int i = blockIdx.x * blockDim.x + threadIdx.x;
    if (i >= total) return;
    int node = i >> 6;
    int d = i & 63;
    const float* gin = gi + (size_t)node * 192;
    const float* ghn = gh + (size_t)node * 192;
    float ir = gin[d], iz = gin[64 + d], in_ = gin[128 + d];
    float hr = ghn[d], hz = ghn[64 + d], hnn = ghn[128 + d];
    float r = 1.0f / (1.0f + expf(-(ir + hr)));
    float z = 1.0f / (1.0f + expf(-(iz + hz)));
    float nn = tanhf(in_ + r * hnn);
    hn[i] = (1.0f - z) * nn + z * h[i];
}

// ---------------------------------------------------------------------------
// Host orchestration (graph-capture safe: no alloc/sync, fixed sequence)
// ---------------------------------------------------------------------------
extern "C" void kernel_launch(void* const* d_in, const int* in_sizes, int n_in,
                              void* d_out, int out_size, void* d_ws, size_t ws_size,
                              hipStream_t stream) {
    (void)in_sizes; (void)n_in; (void)out_size; (void)ws_size;

    constexpr int N = GNN_N, E = GNN_E, D = GNN_D;
    constexpr int N_GCN = 30, N_GGC = 8;

    const float* x         = (const float*)d_in[0];   // [N, 2D]
    const int*   edges     = (const int*)d_in[1];     // [2, E]
    const float* linear_w  = (const float*)d_in[2];   // [D, 2D]
    const float* linear_b  = (const float*)d_in[3];   // [D]
    const float* gcn_w     = (const float*)d_in[4];   // [30, D, D]
    const float* gcn_b     = (const float*)d_in[5];   // [30, D]
    const float* ggc_w     = (const float*)d_in[6];   // [8, D, D]
    const float* gru_w_ih  = (const float*)d_in[7];   // [3D, D]
    const float* gru_w_hh  = (const float*)d_in[8];   // [3D, D]
    const float* gru_b_ih  = (const float*)d_in[9];   // [3D]
    const float* gru_b_hh  = (const float*)d_in[10];  // [3D]
    float* out = (float*)d_out;                       // [N, D]

    const int* src = edges;
    const int* dst = edges + E;

    float* ws = (float*)d_ws;
    float* deg   = ws; ws += N;
    float* dinv  = ws; ws += N;
    float* snorm = ws; ws += N;
    float* norm  = ws; ws += E;
    float* hA    = ws; ws += (size_t)N * D;
    float* hB    = ws; ws += (size_t)N * D;
    float* tmp   = ws; ws += (size_t)N * D;
    float* mbuf  = ws; ws += (size_t)N * D;
    float* gi    = ws; ws += (size_t)N * 3 * D;
    float* gh    = ws; ws += (size_t)N * 3 * D;

    const int T = 256;
    auto blocks = [](long long n, int t) { return (int)((n + t - 1) / t); };

    auto gemm = [&](const float* A, int lda, const float* B, int bks, int bns,
                    const float* bias, float* C, int M, int Nc,
                    int initC, int relu) {
        int tilesN = Nc / 16;
        int groups = (M / 16 + ROW_TILES - 1) / ROW_TILES;
        int waves = groups * tilesN;
        int grid = (waves + (T / 32) - 1) / (T / 32);
        gemm_wmma_k64_kernel<<<grid, T, 0, stream>>>(A, lda, B, bks, bns, bias,
                                                     C, M, Nc, initC, relu);
    };

    // ---- GCN normalization (deg with self-loops, symmetric norm) ----
    deg_init_kernel<<<blocks(N, T), T, 0, stream>>>(deg, N);
    deg_scatter_kernel<<<blocks(E, T), T, 0, stream>>>(dst, deg, E);
    dinv_kernel<<<blocks(N, T), T, 0, stream>>>(deg, dinv, snorm, N);
    edge_norm_kernel<<<blocks(E, T), T, 0, stream>>>(src, dst, dinv, norm, E);

    // ---- initial linear + relu: h = relu(x @ linear_w.T + b) ----
    // split K=128 into two accumulated K=64 passes.
    // B[k][n] = linear_w[n*2D + k]  -> bks=1, bns=2D
    gemm(x,      2 * D, linear_w,      1, 2 * D, linear_b, hA, N, D, 0, 0);
    gemm(x + D,  2 * D, linear_w + D,  1, 2 * D, nullptr,  hA, N, D, 1, 1);

    float* hc = hA;
    float* hn = hB;

    // ---- 30 GCN layers ----
    for (int l = 0; l < N_GCN; ++l) {
        // tmp = h @ W.T  (B[k][n] = W[n*D+k])
        gemm(hc, D, gcn_w + (size_t)l * D * D, 1, D, nullptr, tmp, N, D, 0, 0);
        // hn = tmp*snorm + bias (self-loop term, full overwrite)
        selfloop_bias_kernel<<<blocks((long long)N * D, T), T, 0, stream>>>(
            tmp, snorm, gcn_b + (size_t)l * D, hn, N * D);
        // hn[dst] += tmp[src] * norm
        scatter_edges_kernel<<<blocks((long long)E * 16, T), T, 0, stream>>>(
            tmp, src, dst, norm, hn, E);
        float* t2 = hc; hc = hn; hn = t2;
    }

    // ---- 8 GatedGraphConv steps ----
    for (int s = 0; s < N_GGC; ++s) {
        // tmp = h @ W  (B[k][n] = W[k*D+n])
        gemm(hc, D, ggc_w + (size_t)s * D * D, D, 1, nullptr, tmp, N, D, 0, 0);
        zero_kernel<<<blocks((long long)N * D, T), T, 0, stream>>>(mbuf, N * D);
        scatter_edges_kernel<<<blocks((long long)E * 16, T), T, 0, stream>>>(
            tmp, src, dst, nullptr, mbuf, E);
        // gi = m @ W_ih.T + b_ih ; gh = h @ W_hh.T + b_hh   [N x 3D]
        gemm(mbuf, D, gru_w_ih, 1, D, gru_b_ih, gi, N, 3 * D, 0, 0);
        gemm(hc,   D, gru_w_hh, 1, D, gru_b_hh, gh, N, 3 * D, 0, 0);
        float* tgt = (s == N_GGC - 1) ? out : hn;
        gru_kernel<<<blocks((long long)N * D, T), T, 0, stream>>>(
            gi, gh, hc, tgt, N * D);
        float* old = hc; hc = tgt; hn = old;
    }
}